// Ax_cuda_38259568673510
// MI455X (gfx1250) — compile-verified
//
#include <hip/hip_runtime.h>
#include <stdint.h>

#define B_BATCH 8
#define N_PIX   65536
#define N_ROWS  131072
#define NNZ     16777216

#define WAVES_PER_BLOCK 8
#define BLOCK_THREADS   (WAVES_PER_BLOCK * 32)
#define PIPE   4      // async chunks in flight per wave
#define CHUNK  32     // nnz per chunk (one element per lane)

// ---------------- Kernel 0: transpose X [B,N_PIX] -> Xt [N_PIX,B] ----------------
// Turns the per-nnz gather into two coalesced 16B loads from one 32B segment.
__global__ void transpose_X_kernel(const float* __restrict__ X, float* __restrict__ Xt) {
  int c = blockIdx.x * blockDim.x + threadIdx.x;
  if (c >= N_PIX) return;
  float4 lo, hi;
  lo.x = X[0 * N_PIX + c]; lo.y = X[1 * N_PIX + c];
  lo.z = X[2 * N_PIX + c]; lo.w = X[3 * N_PIX + c];
  hi.x = X[4 * N_PIX + c]; hi.y = X[5 * N_PIX + c];
  hi.z = X[6 * N_PIX + c]; hi.w = X[7 * N_PIX + c];
  float4* dst = (float4*)Xt;
  dst[2 * c + 0] = lo;
  dst[2 * c + 1] = hi;
}

// ---------------- Kernel 1: COO (sorted rows) -> CSR row_ptr via binary search ----
__global__ void build_row_ptr_kernel(const int* __restrict__ rows, int* __restrict__ row_ptr) {
  int r = blockIdx.x * blockDim.x + threadIdx.x;
  if (r > N_ROWS) return;
  int lo = 0, hi = NNZ;             // lower_bound: first idx with rows[idx] >= r
  while (lo < hi) {
    int mid = (lo + hi) >> 1;
    if (rows[mid] < r) lo = mid + 1; else hi = mid;
  }
  row_ptr[r] = lo;
}

// ---------------- Kernel 2: one wave32 per ray; async-LDS streamed COO ------------
template <bool XT>
__global__ __launch_bounds__(BLOCK_THREADS)
void spmm_rows_kernel(const float* __restrict__ Xsrc,
                      const float* __restrict__ vals,
                      const int* __restrict__ cols,
                      const int* __restrict__ row_ptr,
                      float* __restrict__ out) {
  __shared__ float s_vals[WAVES_PER_BLOCK][PIPE][CHUNK];
  __shared__ int   s_cols[WAVES_PER_BLOCK][PIPE][CHUNK];

  const int lane = threadIdx.x & 31;
  const int wv   = threadIdx.x >> 5;
  const int row  = blockIdx.x * WAVES_PER_BLOCK + wv;
  const int start = row_ptr[row];
  const int end   = row_ptr[row + 1];
  const int nchunks = (end - start + CHUNK - 1) / CHUNK;

  float acc[B_BATCH];
#pragma unroll
  for (int b = 0; b < B_BATCH; ++b) acc[b] = 0.0f;

#if defined(__HIP_DEVICE_COMPILE__)
  // LDS byte addresses: generic shared-pointer low 32 bits == LDS offset (ISA 10.2).
  const uint32_t lv_base = (uint32_t)(uintptr_t)&s_vals[wv][0][lane];
  const uint32_t lc_base = (uint32_t)(uintptr_t)&s_cols[wv][0][lane];

  auto issue = [&](int c) {
    int gi = start + c * CHUNK + lane;
    gi = gi < (NNZ - 1) ? gi : (NNZ - 1);            // clamp; masked at consume
    uint32_t lv = lv_base + (uint32_t)((c & (PIPE - 1)) * (CHUNK * 4));
    uint32_t lc = lc_base + (uint32_t)((c & (PIPE - 1)) * (CHUNK * 4));
    uint64_t gv = (uint64_t)(uintptr_t)(vals + gi);
    uint64_t gc = (uint64_t)(uintptr_t)(cols + gi);
    asm volatile("global_load_async_to_lds_b32 %0, %1, off" :: "v"(lv), "v"(gv) : "memory");
    asm volatile("global_load_async_to_lds_b32 %0, %1, off" :: "v"(lc), "v"(gc) : "memory");
  };

  {
    int pre = nchunks < (PIPE - 1) ? nchunks : (PIPE - 1);
    for (int c = 0; c < pre; ++c) issue(c);          // prime the pipeline
  }

  for (int c = 0; c < nchunks; ++c) {
    // Chunks issued beyond c so far: up to PIPE-2. Wait so chunk c's 2 ops retired.
    int ahead = nchunks - 1 - c;
    if (ahead > PIPE - 2) ahead = PIPE - 2;
    if (ahead >= 2)      asm volatile("s_wait_asynccnt 4" ::: "memory");
    else if (ahead == 1) asm volatile("s_wait_asynccnt 2" ::: "memory");
    else                 asm volatile("s_wait_asynccnt 0" ::: "memory");

    const int  idx  = start + c * CHUNK + lane;
    const bool live = idx < end;
    float v = 0.0f;
    int   col = 0;
    if (live) {
      v   = s_vals[wv][c & (PIPE - 1)][lane];
      col = s_cols[wv][c & (PIPE - 1)][lane];
    }
    // WAR safety: LDS reads retired before this slot's buffer is refilled.
    asm volatile("s_wait_dscnt 0" ::: "memory");
    if (c + PIPE - 1 < nchunks) issue(c + PIPE - 1);

    if (live) {
      if (XT) {
        const float4* xt = (const float4*)Xsrc;      // L2-resident (2 MB of 192 MB)
        float4 lo = xt[2 * (size_t)col + 0];
        float4 hi = xt[2 * (size_t)col + 1];
        acc[0] = fmaf(v, lo.x, acc[0]); acc[1] = fmaf(v, lo.y, acc[1]);
        acc[2] = fmaf(v, lo.z, acc[2]); acc[3] = fmaf(v, lo.w, acc[3]);
        acc[4] = fmaf(v, hi.x, acc[4]); acc[5] = fmaf(v, hi.y, acc[5]);
        acc[6] = fmaf(v, hi.z, acc[6]); acc[7] = fmaf(v, hi.w, acc[7]);
      } else {
#pragma unroll
        for (int b = 0; b < B_BATCH; ++b)
          acc[b] = fmaf(v, Xsrc[(size_t)b * N_PIX + col], acc[b]);
      }
    }
  }
#else
  // Host-pass (stub) body: plain loads, same semantics.
  for (int i = start + lane; i < end; i += 32) {
    float v = vals[i];
    int   col = cols[i];
    if (XT) {
      const float4* xt = (const float4*)Xsrc;
      float4 lo = xt[2 * (size_t)col + 0];
      float4 hi = xt[2 * (size_t)col + 1];
      acc[0] += v * lo.x; acc[1] += v * lo.y; acc[2] += v * lo.z; acc[3] += v * lo.w;
      acc[4] += v * hi.x; acc[5] += v * hi.y; acc[6] += v * hi.z; acc[7] += v * hi.w;
    } else {
#pragma unroll
      for (int b = 0; b < B_BATCH; ++b) acc[b] += v * Xsrc[(size_t)b * N_PIX + col];
    }
  }
  (void)s_vals; (void)s_cols;
#endif

  // wave32 segmented reduction: 5 xor-shuffle steps per batch slot
#pragma unroll
  for (int b = 0; b < B_BATCH; ++b) {
    float s = acc[b];
#pragma unroll
    for (int off = 16; off > 0; off >>= 1) s += __shfl_xor(s, off, 32);
    acc[b] = s;
  }
  if (lane == 0) {
#pragma unroll
    for (int b = 0; b < B_BATCH; ++b)
      out[(size_t)b * N_ROWS + row] = acc[b];        // Y.T layout [B, N_ROWS]
  }
}

// ---------------- Launch ----------------
extern "C" void kernel_launch(void* const* d_in, const int* in_sizes, int n_in,
                              void* d_out, int out_size, void* d_ws, size_t ws_size,
                              hipStream_t stream) {
  const float* X    = (const float*)d_in[0];
  const float* vals = (const float*)d_in[1];
  const int*   rows = (const int*)d_in[2];
  const int*   cols = (const int*)d_in[3];
  float* out = (float*)d_out;

  // Workspace layout: [0, (N_ROWS+1)*4) row_ptr ; [1MB, 1MB+2MB) Xt
  int* row_ptr = (int*)d_ws;
  const size_t XT_OFF   = (size_t)1 << 20;
  const size_t XT_BYTES = (size_t)N_PIX * B_BATCH * sizeof(float);
  const bool use_xt = ws_size >= XT_OFF + XT_BYTES;

  build_row_ptr_kernel<<<(N_ROWS + 1 + 255) / 256, 256, 0, stream>>>(rows, row_ptr);

  if (use_xt) {
    float* Xt = (float*)((char*)d_ws + XT_OFF);
    transpose_X_kernel<<<N_PIX / 256, 256, 0, stream>>>(X, Xt);
    spmm_rows_kernel<true><<<N_ROWS / WAVES_PER_BLOCK, BLOCK_THREADS, 0, stream>>>(
        Xt, vals, cols, row_ptr, out);
  } else {
    spmm_rows_kernel<false><<<N_ROWS / WAVES_PER_BLOCK, BLOCK_THREADS, 0, stream>>>(
        X, vals, cols, row_ptr, out);
  }
}